// PromptLearnerForBERT_61409442398689
// MI455X (gfx1250) — compile-verified
//
#include <hip/hip_runtime.h>
#include <hip/hip_bf16.h>

// PromptLearnerForBERT: masked scatter of prompt embeddings into pad slots.
// B=32, L=512, D=768, N_PROMPT=20, N_SAMPLES=10000.
// Pure data movement (~103 MB, 0 FLOP) -> HBM-bound (~4.4us @ 23.3 TB/s).
// Strategy: tiny wave32-ballot scan kernel to compute per-position source row,
// then a bulk streaming kernel using the CDNA5 async global<->LDS DMA path
// (GLOBAL_LOAD_ASYNC_TO_LDS_B128 / GLOBAL_STORE_ASYNC_FROM_LDS_B128,
//  ASYNCcnt-tracked) so data never round-trips through VGPRs.

#define kB 32
#define kL 512
#define kD 768              // floats per row = 3072 bytes = 192 x 16B
#define kNP 20

typedef __attribute__((address_space(3))) unsigned char lds_uchar_t;

// ---------------------------------------------------------------------------
// Kernel 1: per-row pad-rank scan (wave32 ballot + cross-wave LDS combine).
// One 512-thread block per batch row b. Writes:
//   src_row[b*L+l] = sample_id[b]*20 + rank   if this slot takes a prompt row
//                  = -1                        otherwise (pass-through copy)
//   out_mask[b*L+l] = 1 if taken else original mask (int32 raw bits)
// ---------------------------------------------------------------------------
__global__ __launch_bounds__(kL) void prompt_rank_kernel(
    const int* __restrict__ sample_ids,
    const int* __restrict__ attn_mask,
    int* __restrict__ src_row,
    int* __restrict__ out_mask)
{
    __shared__ int wave_tot[kL / 32];

    const int b = blockIdx.x;
    const int l = threadIdx.x;
    const int idx = b * kL + l;

    const int m = attn_mask[idx];
    const bool pad = (m == 0);

    const unsigned lane = (unsigned)(l & 31);
    const unsigned wave = (unsigned)(l >> 5);

    // wave32 ballot of pad flags
    unsigned bal = __builtin_amdgcn_ballot_w32(pad);
    const int prefix_in_wave =
        __popc(bal & (unsigned)((1ull << lane) - 1ull));   // pads strictly before me
    const int wave_total = __popc(bal);

    if (lane == 0) wave_tot[wave] = wave_total;
    __syncthreads();

    int base = 0;
    #pragma unroll
    for (unsigned w = 0; w < kL / 32; ++w)
        base += (w < wave) ? wave_tot[w] : 0;

    // rank = cumsum(pad)[l] - 1 == (# pads strictly before l) when pad[l]==1
    const int rank = base + prefix_in_wave;
    const bool take = pad && (rank < kNP);

    const int sid = sample_ids[b];
    src_row[idx] = take ? (sid * kNP + rank) : -1;
    out_mask[idx] = take ? 1 : m;
}

// ---------------------------------------------------------------------------
// Kernel 2: bulk stream. One block per (b,l) position, 192 threads, each
// thread DMAs one 16-byte chunk of the 3072-byte embedding row through LDS
// using the CDNA5 async global<->LDS path. The source-select branch is
// uniform across the block (scalar branch, no divergence).
// ---------------------------------------------------------------------------
__global__ __launch_bounds__(kD / 4) void prompt_stream_kernel(
    const float* __restrict__ input_embeds,
    const float* __restrict__ prompts,
    const int* __restrict__ src_row,
    float* __restrict__ out)
{
    __shared__ __align__(16) unsigned char stage[(kD / 4) * 16];  // 3072 B

    const unsigned row = blockIdx.x;            // row = b*kL + l
    const unsigned t = threadIdx.x;
    const unsigned byte_off = t * 16u;

    const int pr = src_row[row];                // uniform per block
    const unsigned char* src =
        (pr >= 0) ? (const unsigned char*)(prompts + (size_t)pr * kD)
                  : (const unsigned char*)(input_embeds + (size_t)row * kD);
    unsigned char* dst = (unsigned char*)(out + (size_t)row * kD);

    // 32-bit workgroup-relative LDS byte offset for the async ops
    const unsigned lds_off = (unsigned)(size_t)(lds_uchar_t*)&stage[byte_off];

    const unsigned long long gsrc = (unsigned long long)(src + byte_off);
    const unsigned long long gdst = (unsigned long long)(dst + byte_off);

    // memory -> LDS (ASYNCcnt++)
    asm volatile("global_load_async_to_lds_b128 %0, %1, off"
                 :: "v"(lds_off), "v"(gsrc)
                 : "memory");
    // wait for this wave's async loads to land in LDS
    asm volatile("s_wait_asynccnt 0" ::: "memory");
    // LDS -> memory (ASYNCcnt++); s_endpgm's implicit wait-idle drains it
    asm volatile("global_store_async_from_lds_b128 %0, %1, off"
                 :: "v"(gdst), "v"(lds_off)
                 : "memory");
}

// ---------------------------------------------------------------------------
// Host launcher.
// Inputs (setup_inputs order):
//   d_in[0] sample_id_tensor [32]        int
//   d_in[1] input_embeds     [32,512,768] f32
//   d_in[2] attention_mask   [32,512]    int
//   d_in[3] prompt_embeddings[10000,20,768] f32
// Output: embeds (32*512*768 f32) then mask (32*512 int32 raw) concatenated.
// ---------------------------------------------------------------------------
extern "C" void kernel_launch(void* const* d_in, const int* in_sizes, int n_in,
                              void* d_out, int out_size, void* d_ws, size_t ws_size,
                              hipStream_t stream) {
    const int*   sample_ids   = (const int*)d_in[0];
    const float* input_embeds = (const float*)d_in[1];
    const int*   attn_mask    = (const int*)d_in[2];
    const float* prompts      = (const float*)d_in[3];

    float* out_embeds = (float*)d_out;
    int*   out_mask   = (int*)(out_embeds + (size_t)kB * kL * kD);
    int*   src_row    = (int*)d_ws;   // kB*kL ints = 64 KB scratch

    prompt_rank_kernel<<<kB, kL, 0, stream>>>(sample_ids, attn_mask,
                                              src_row, out_mask);
    prompt_stream_kernel<<<kB * kL, kD / 4, 0, stream>>>(input_embeds, prompts,
                                                         src_row, out_embeds);
}